// FactorizationMachine_23931557773661
// MI455X (gfx1250) — compile-verified
//
#include <hip/hip_runtime.h>
#include <hip/hip_bf16.h>

typedef __attribute__((ext_vector_type(2))) float v2f;
typedef __attribute__((ext_vector_type(8))) float v8f;

#define KDIM 16

__device__ __forceinline__ int lower_bound_i32(const int* __restrict__ arr, int n, int key) {
    int lo = 0, hi = n;
    while (lo < hi) {
        int mid = (lo + hi) >> 1;
        if (arr[mid] < key) lo = mid + 1; else hi = mid;
    }
    return lo;
}

// Branch-free clamped load of one 4-nonzero slot's streaming data.
// Caller guarantees hi-1 >= 0. Out-of-segment lanes read a valid address
// (hi-1) and have their value zeroed by v_cndmask (no exec-mask branches).
__device__ __forceinline__ void load_slot(const int*   __restrict__ feats,
                                          const float* __restrict__ values,
                                          int jbase, int k0, int hi,
                                          int& fa, int& fb, float& va, float& vb) {
    const int ja  = jbase + k0;
    const int jb  = ja + 1;
    const bool oka = (ja < hi);
    const bool okb = (jb < hi);
    const int jac = oka ? ja : (hi - 1);
    const int jbc = okb ? jb : (hi - 1);
    const float var = values[jac];
    const float vbr = values[jbc];
    fa = feats[jac];
    fb = feats[jbc];
    va = oka ? var : 0.f;
    vb = okb ? vbr : 0.f;
}

// One wave32 per batch segment. Segment MACs run on the matrix pipe:
//   C1 += A(v)   x B(e)     -> t1[n] replicated across rows of C
//   C2 += A(v^2) x B(e^2)   -> t2[n]
// A-matrix (16x4 f32, 2 VGPRs): lane L, vgpr p holds A[M=L%16][K=(L>=16?2:0)+p]
//   -> values[jbase + k], independent of M (row-replicated down M).
// B-matrix (4x16 f32, 2 VGPRs): lane L, vgpr p holds B[K=(L>=16?2:0)+p][N=L%16]
//   -> embedding[feats[jbase+k]*16 + (L%16)]; 16 lanes cover one 64B row.
// The streaming (feats/values) loads are software-pipelined one iteration
// ahead so they overlap the dependent embedding gathers + WMMAs.
__global__ __launch_bounds__(256)
void fm_fused_kernel(const int*   __restrict__ index,
                     const int*   __restrict__ feats,
                     const float* __restrict__ values,
                     const float* __restrict__ bias,
                     const float* __restrict__ weights,
                     const float* __restrict__ embedding,
                     float*       __restrict__ out,
                     int batch, int nnz)
{
    const int lane = (int)(threadIdx.x & 31);
    const int wave = (int)(threadIdx.x >> 5);
    const int b = (int)blockIdx.x * ((int)blockDim.x >> 5) + wave;
    if (b >= batch) return;                       // wave-uniform

    // Segment bounds: index[] is sorted; all lanes search redundantly (uniform).
    const int lo = lower_bound_i32(index, nnz, b);
    const int hi = lower_bound_i32(index, nnz, b + 1);

    const int n  = lane & 15;                     // N column / embedding k-slot
    const int k0 = (lane >> 4) << 1;              // K base: 0 for lanes 0-15, 2 for 16-31

    v8f c1 = {};                                  // t1 accumulator (matrix pipe)
    v8f c2 = {};                                  // t2 accumulator (matrix pipe)
    float accw = 0.f;                             // first-order partial

    const int iters = (hi - lo + 3) >> 2;         // wave-uniform trip count
    if (iters > 0) {                              // uniform: hi-1 >= lo is valid
        int   fa, fb;
        float va, vb;
        load_slot(feats, values, lo, k0, hi, fa, fb, va, vb);   // prologue

        for (int it = 0; it < iters; ++it) {
            // Dependent gathers for the current slot.
            const float ea = embedding[fa * KDIM + n];
            const float eb = embedding[fb * KDIM + n];

            // Issue next slot's independent streaming loads while the
            // embedding gathers are in flight (indices self-clamp past hi).
            int   fna, fnb;
            float vna, vnb;
            load_slot(feats, values, lo + ((it + 1) << 2), k0, hi,
                      fna, fnb, vna, vnb);

            // First-order: lanes with n==0 (L=0 covers slots 0,1; L=16 covers
            // 2,3) accumulate w*v; compiler predicates the 2-lane weight loads.
            accw += (n == 0) ? (weights[fa] * va + weights[fb] * vb) : 0.f;

            v2f A1; A1.x = va;      A1.y = vb;
            v2f B1; B1.x = ea;      B1.y = eb;
            v2f A2; A2.x = va * va; A2.y = vb * vb;
            v2f B2; B2.x = ea * ea; B2.y = eb * eb;

            c1 = __builtin_amdgcn_wmma_f32_16x16x4_f32(
                     false, A1, false, B1, (short)0, c1, false, false);
            c2 = __builtin_amdgcn_wmma_f32_16x16x4_f32(
                     false, A2, false, B2, (short)0, c2, false, false);

            // Rotate pipeline registers.
            fa = fna; fb = fnb; va = vna; vb = vnb;
        }
    }

    // Row 0 of C lives in c[0]: lane L holds t{1,2}[L&15] (rows replicated,
    // lanes 16-31 hold row M=8 which is identical). Mask the upper half to
    // avoid double counting.
    const float t1 = c1[0];
    const float t2 = c2[0];
    float part = (lane < 16) ? 0.5f * (t1 * t1 - t2) : 0.f;

    // Full wave32 xor-reduction of both scalars (ds_bpermute path).
    #pragma unroll
    for (int m = 16; m >= 1; m >>= 1) {
        part += __shfl_xor(part, m, 32);
        accw += __shfl_xor(accw, m, 32);
    }

    if (lane == 0) {
        const float x = accw + bias[0] + part;
        out[b] = 1.f / (1.f + __expf(-x));
    }
}

extern "C" void kernel_launch(void* const* d_in, const int* in_sizes, int n_in,
                              void* d_out, int out_size, void* d_ws, size_t ws_size,
                              hipStream_t stream) {
    // setup_inputs order:
    //   0: batch_size (scalar), 1: index (nnz), 2: feats (nnz), 3: values (nnz),
    //   4: bias (1), 5: weights (VOCAB), 6: embedding (VOCAB*16)
    const int*   index     = (const int*)  d_in[1];
    const int*   feats     = (const int*)  d_in[2];
    const float* values    = (const float*)d_in[3];
    const float* bias      = (const float*)d_in[4];
    const float* weights   = (const float*)d_in[5];
    const float* embedding = (const float*)d_in[6];
    float* out = (float*)d_out;

    const int nnz   = in_sizes[1];
    const int batch = out_size;          // output is (1, batch)

    const int wavesPerBlock = 8;         // 256 threads = 8 wave32
    dim3 block(32 * wavesPerBlock);
    dim3 grid((batch + wavesPerBlock - 1) / wavesPerBlock);
    fm_fused_kernel<<<grid, block, 0, stream>>>(index, feats, values, bias,
                                                weights, embedding, out,
                                                batch, nnz);
}